// GraphEmbedder3_64372969832921
// MI455X (gfx1250) — compile-verified
//
#include <hip/hip_runtime.h>
#include <math.h>

#define NNODES 10000
#define NEDGES 160000
#define NGRAPH 64

typedef __attribute__((ext_vector_type(2))) float v2f;
typedef __attribute__((ext_vector_type(8))) float v8f;

#define ENC_NEG_INF 0x007FFFFFu

__device__ __forceinline__ float elu_f(float x) { return x > 0.f ? x : (expf(x) - 1.f); }

// order-preserving float <-> uint encoding for atomicMax on floats
__device__ __forceinline__ unsigned fenc(float f) {
  unsigned u = __float_as_uint(f);
  return (u & 0x80000000u) ? ~u : (u | 0x80000000u);
}
__device__ __forceinline__ float fdec(unsigned u) {
  return __uint_as_float((u & 0x80000000u) ? (u & 0x7fffffffu) : ~u);
}

// ---------------------------------------------------------------------------
// FP32 WMMA GEMM:  Y[N,O] = X[N,K] @ W[O,K]^T + bias
// block = 256 threads = 8 waves; tile BM=64 x BN=32; BK=32
// each wave computes one 16x16 tile with v_wmma_f32_16x16x4_f32
// ---------------------------------------------------------------------------
#define BM 64
#define BN 32
#define BK 32

__global__ __launch_bounds__(256) void gemm_wmma_f32(
    const float* __restrict__ X, const float* __restrict__ W,
    const float* __restrict__ bias, float* __restrict__ Y,
    int N, int K, int O) {
  __shared__ float Xs[BM][BK + 1];
  __shared__ float Ws[BN][BK + 1];

  const int t = threadIdx.x;
  const int m0 = blockIdx.x * BM;
  const int n0 = blockIdx.y * BN;
  const int wave = t >> 5;
  const int lane = t & 31;
  const int wm = (wave >> 1) * 16;  // 0,16,32,48
  const int wn = (wave & 1) * 16;   // 0,16
  const int lr = lane & 15;         // m/n within tile
  const int lh = lane >> 4;         // K-half selector

  v8f acc = {0.f, 0.f, 0.f, 0.f, 0.f, 0.f, 0.f, 0.f};

  const int col = t & 31;
  const int rowb = t >> 5;  // 0..7

  for (int k0 = 0; k0 < K; k0 += BK) {
#pragma unroll
    for (int i = 0; i < 8; ++i) {  // 64x32 X tile
      int r = rowb + i * 8;
      int gm = m0 + r, gk = k0 + col;
      Xs[r][col] = (gm < N && gk < K) ? X[(size_t)gm * K + gk] : 0.f;
    }
#pragma unroll
    for (int i = 0; i < 4; ++i) {  // 32x32 W tile
      int r = rowb + i * 8;
      int gn = n0 + r, gk = k0 + col;
      Ws[r][col] = (gn < O && gk < K) ? W[(size_t)gn * K + gk] : 0.f;
    }
    __syncthreads();
#pragma unroll
    for (int kk = 0; kk < BK; kk += 4) {
      v2f a, b;
      // A 16x4 fp32 layout: lanes 0-15 hold K=0,1 (v0,v1); lanes 16-31 hold K=2,3
      a.x = Xs[wm + lr][kk + 2 * lh + 0];
      a.y = Xs[wm + lr][kk + 2 * lh + 1];
      // B 4x16 fp32 layout mirrors A with n = lane%16
      b.x = Ws[wn + lr][kk + 2 * lh + 0];
      b.y = Ws[wn + lr][kk + 2 * lh + 1];
      acc = __builtin_amdgcn_wmma_f32_16x16x4_f32(false, a, false, b,
                                                  (short)0, acc, false, false);
    }
    __syncthreads();
  }

  // C/D layout: vgpr v, lane -> m = v + 8*(lane/16), n = lane%16
#pragma unroll
  for (int v = 0; v < 8; ++v) {
    int gm = m0 + wm + v + 8 * lh;
    int gn = n0 + wn + lr;
    if (gm < N && gn < O) {
      float bv = bias ? bias[gn] : 0.f;
      Y[(size_t)gm * O + gn] = acc[v] + bv;
    }
  }
}

// ---------------------------------------------------------------------------
// small utility kernels
// ---------------------------------------------------------------------------
__global__ void kzero(float* p, int n) {
  int i = blockIdx.x * blockDim.x + threadIdx.x;
  if (i < n) p[i] = 0.f;
}

__global__ void init_softmax(unsigned* nmax, float* nsum, int n) {
  int i = blockIdx.x * blockDim.x + threadIdx.x;
  if (i < n) { nmax[i] = ENC_NEG_INF; nsum[i] = 0.f; }
}

// y[i] = (elu?) (y[i] + bias[i % D])
__global__ void bias_act(float* y, const float* __restrict__ bias, int total,
                         int D, int do_elu) {
  int i = blockIdx.x * blockDim.x + threadIdx.x;
  if (i >= total) return;
  float v = y[i] + bias[i % D];
  if (do_elu) v = elu_f(v);
  y[i] = v;
}

// out = xg + elu(tr) + (res ? res : 0)
__global__ void combine_k(const float* __restrict__ xg,
                          const float* __restrict__ tr,
                          const float* __restrict__ res, float* out, int total) {
  int i = blockIdx.x * blockDim.x + threadIdx.x;
  if (i >= total) return;
  float v = xg[i] + elu_f(tr[i]);
  if (res) v += res[i];
  out[i] = v;
}

// out = elu(a) + b
__global__ void elu_add_k(const float* __restrict__ a,
                          const float* __restrict__ b, float* out, int total) {
  int i = blockIdx.x * blockDim.x + threadIdx.x;
  if (i >= total) return;
  out[i] = elu_f(a[i]) + b[i];
}

// ---------------------------------------------------------------------------
// GraphNorm (3 passes, per-graph atomics)
// ---------------------------------------------------------------------------
__global__ void gn_sum(const float* __restrict__ x, const int* __restrict__ batch,
                       float* gsum, float* cnt, int n, int D) {
  int i = blockIdx.x * blockDim.x + threadIdx.x;
  if (i >= n * D) return;
  int node = i / D, j = i % D;
  int g = batch[node];
  atomicAdd(gsum + g * D + j, x[i]);
  if (j == 0) atomicAdd(cnt + g, 1.f);
}

__global__ void gn_center(const float* __restrict__ x,
                          const int* __restrict__ batch,
                          const float* __restrict__ gsum,
                          const float* __restrict__ cnt,
                          const float* __restrict__ ms, float* xc, float* gvar,
                          int n, int D) {
  int i = blockIdx.x * blockDim.x + threadIdx.x;
  if (i >= n * D) return;
  int node = i / D, j = i % D;
  int g = batch[node];
  float mean = gsum[g * D + j] / cnt[g];
  float v = x[i] - ms[j] * mean;
  xc[i] = v;
  atomicAdd(gvar + g * D + j, v * v);
}

__global__ void gn_norm(float* xc, const int* __restrict__ batch,
                        const float* __restrict__ gvar,
                        const float* __restrict__ cnt,
                        const float* __restrict__ w, const float* __restrict__ b,
                        int n, int D, int do_elu) {
  int i = blockIdx.x * blockDim.x + threadIdx.x;
  if (i >= n * D) return;
  int node = i / D, j = i % D;
  int g = batch[node];
  float var = gvar[g * D + j] / cnt[g];
  float o = w[j] * xc[i] / sqrtf(var + 1e-5f) + b[j];
  if (do_elu) o = elu_f(o);
  xc[i] = o;
}

// ---------------------------------------------------------------------------
// LayerNorm: one wave32 per row
// ---------------------------------------------------------------------------
__global__ __launch_bounds__(256) void ln_k(const float* __restrict__ x,
                                            const float* __restrict__ w,
                                            const float* __restrict__ b,
                                            float* y, int N, int D, float eps) {
  int row = blockIdx.x * (blockDim.x >> 5) + (threadIdx.x >> 5);
  int lane = threadIdx.x & 31;
  if (row >= N) return;
  const float* xr = x + (size_t)row * D;
  float s = 0.f;
  for (int j = lane; j < D; j += 32) s += xr[j];
#pragma unroll
  for (int off = 16; off; off >>= 1) s += __shfl_xor(s, off);
  float mu = s / D;
  float v = 0.f;
  for (int j = lane; j < D; j += 32) { float t = xr[j] - mu; v += t * t; }
#pragma unroll
  for (int off = 16; off; off >>= 1) v += __shfl_xor(v, off);
  float inv = 1.f / sqrtf(v / D + eps);
  for (int j = lane; j < D; j += 32)
    y[(size_t)row * D + j] = (xr[j] - mu) * inv * w[j] + b[j];
}

// ---------------------------------------------------------------------------
// GATv2 edge kernels
// ---------------------------------------------------------------------------
__global__ void gat_scores(const float* __restrict__ xl,
                           const float* __restrict__ xr,
                           const float* __restrict__ ea,
                           const float* __restrict__ We,
                           const float* __restrict__ att,
                           const int* __restrict__ src,
                           const int* __restrict__ dst, float* sbuf,
                           unsigned* nmax, int E, int H, int C) {
  int i = blockIdx.x * blockDim.x + threadIdx.x;
  if (i >= E * H) return;
  int e = i / H, h = i % H;
  int s = src[e], d = dst[e];
  float a0 = ea[e * 3 + 0], a1 = ea[e * 3 + 1], a2 = ea[e * 3 + 2];
  int HC = H * C;
  const float* pl = xl + (size_t)s * HC + h * C;
  const float* pr = xr + (size_t)d * HC + h * C;
  const float* pw = We + (size_t)h * C * 3;
  const float* pa = att + h * C;
  float acc = 0.f;
  for (int c = 0; c < C; ++c) {
    float ep = pw[c * 3 + 0] * a0 + pw[c * 3 + 1] * a1 + pw[c * 3 + 2] * a2;
    float m = pl[c] + pr[c] + ep;
    m = m > 0.f ? m : 0.2f * m;  // leaky_relu 0.2
    acc += pa[c] * m;
  }
  sbuf[i] = acc;
  atomicMax(nmax + (size_t)d * H + h, fenc(acc));
}

__global__ void seg_exp(const int* __restrict__ dst, float* sbuf,
                        const unsigned* __restrict__ nmax, float* nsum, int E,
                        int H) {
  int i = blockIdx.x * blockDim.x + threadIdx.x;
  if (i >= E * H) return;
  int e = i / H, h = i % H;
  int d = dst[e];
  float w = expf(sbuf[i] - fdec(nmax[(size_t)d * H + h]));
  sbuf[i] = w;
  atomicAdd(nsum + (size_t)d * H + h, w);
}

__global__ void gat_msg(const float* __restrict__ xl,
                        const int* __restrict__ src,
                        const int* __restrict__ dst,
                        const float* __restrict__ sbuf,
                        const float* __restrict__ nsum, float* out, int E,
                        int H, int C) {
  int HC = H * C;
  int i = blockIdx.x * blockDim.x + threadIdx.x;
  if (i >= E * HC) return;
  int e = i / HC, hc = i % HC, h = hc / C;
  int d = dst[e];
  float alpha = sbuf[e * H + h] / (nsum[(size_t)d * H + h] + 1e-16f);
  atomicAdd(out + (size_t)d * HC + hc, xl[(size_t)src[e] * HC + hc] * alpha);
}

// ---------------------------------------------------------------------------
// Transformer-conv edge kernels
// ---------------------------------------------------------------------------
__global__ void trans_scores(const float* __restrict__ q,
                             const float* __restrict__ k,
                             const float* __restrict__ ea,
                             const float* __restrict__ We,
                             const float* __restrict__ be,
                             const int* __restrict__ src,
                             const int* __restrict__ dst, float* sbuf,
                             unsigned* nmax, int E, int H, int C, float scale) {
  int i = blockIdx.x * blockDim.x + threadIdx.x;
  if (i >= E * H) return;
  int e = i / H, h = i % H;
  int s = src[e], d = dst[e];
  float a0 = ea[e * 3 + 0], a1 = ea[e * 3 + 1], a2 = ea[e * 3 + 2];
  int HC = H * C;
  const float* pq = q + (size_t)d * HC + h * C;
  const float* pk = k + (size_t)s * HC + h * C;
  const float* pw = We + (size_t)h * C * 3;
  const float* pb = be + h * C;
  float acc = 0.f;
  for (int c = 0; c < C; ++c) {
    float ep = pw[c * 3 + 0] * a0 + pw[c * 3 + 1] * a1 + pw[c * 3 + 2] * a2 + pb[c];
    acc += pq[c] * (pk[c] + ep);
  }
  acc *= scale;
  sbuf[i] = acc;
  atomicMax(nmax + (size_t)d * H + h, fenc(acc));
}

__global__ void trans_msg(const float* __restrict__ v,
                          const float* __restrict__ ea,
                          const float* __restrict__ We,
                          const float* __restrict__ be,
                          const int* __restrict__ src,
                          const int* __restrict__ dst,
                          const float* __restrict__ sbuf,
                          const float* __restrict__ nsum, float* out, int E,
                          int H, int C) {
  int HC = H * C;
  int i = blockIdx.x * blockDim.x + threadIdx.x;
  if (i >= E * HC) return;
  int e = i / HC, hc = i % HC, h = hc / C;
  int d = dst[e];
  float alpha = sbuf[e * H + h] / (nsum[(size_t)d * H + h] + 1e-16f);
  float a0 = ea[e * 3 + 0], a1 = ea[e * 3 + 1], a2 = ea[e * 3 + 2];
  float ep = We[hc * 3 + 0] * a0 + We[hc * 3 + 1] * a1 + We[hc * 3 + 2] * a2 + be[hc];
  float val = (v[(size_t)src[e] * HC + hc] + ep) * alpha;
  atomicAdd(out + (size_t)d * HC + hc, val);
}

// ---------------------------------------------------------------------------
// pooling + final head
// ---------------------------------------------------------------------------
__global__ void pool_k(const float* __restrict__ x,
                       const int* __restrict__ batch, float* pooled, int N,
                       int D) {
  int i = blockIdx.x * blockDim.x + threadIdx.x;
  if (i >= N * D) return;
  int n = i / D, j = i % D;
  atomicAdd(pooled + batch[n] * D + j, x[i]);
}

__global__ void final_k(const float* __restrict__ pooled,
                        const float* __restrict__ W,
                        const float* __restrict__ bw,
                        const float* __restrict__ lw,
                        const float* __restrict__ lb, float* out) {
  int g = threadIdx.x;
  if (g >= NGRAPH) return;
  float y[18];
  float mu = 0.f;
  for (int j = 0; j < 18; ++j) {
    float a = bw[j];
    for (int k = 0; k < 18; ++k) a += pooled[g * 18 + k] * W[j * 18 + k];
    y[j] = a;
    mu += a;
  }
  mu /= 18.f;
  float var = 0.f;
  for (int j = 0; j < 18; ++j) { float t = y[j] - mu; var += t * t; }
  var /= 18.f;
  float inv = 1.f / sqrtf(var + 1e-5f);
  for (int j = 0; j < 18; ++j) {
    float z = (y[j] - mu) * inv * lw[j] + lb[j];
    out[g * 18 + j] = elu_f(z);
  }
}

// ---------------------------------------------------------------------------
// host orchestration
// ---------------------------------------------------------------------------
static inline int ceilDiv(int a, int b) { return (a + b - 1) / b; }

extern "C" void kernel_launch(void* const* d_in, const int* in_sizes, int n_in,
                              void* d_out, int out_size, void* d_ws,
                              size_t ws_size, hipStream_t stream) {
  (void)in_sizes; (void)n_in; (void)out_size; (void)ws_size;
  const int N = NNODES, E = NEDGES, G = NGRAPH;

  const float* x_in = (const float*)d_in[0];
  const int* src = (const int*)d_in[1];
  const int* dst = src + E;
  const float* ea = (const float*)d_in[2];
  const int* batch = (const int*)d_in[3];
  auto P = [&](int i) { return (const float*)d_in[i]; };

  // workspace layout (floats)
  const size_t SLOT = (size_t)N * 288;
  float* ws = (float*)d_ws;
  float* S0 = ws + 0 * SLOT;  // current x / residual
  float* S1 = ws + 1 * SLOT;  // xn / post-attention x
  float* S2 = ws + 2 * SLOT;  // xl,q,v / x_ln
  float* S3 = ws + 3 * SLOT;  // xr,k / lin out
  float* S4 = ws + 4 * SLOT;  // gat accumulator -> xg
  float* S5 = ws + 5 * SLOT;  // trans accumulator (skip + messages)
  float* sbuf = ws + 6 * SLOT;                 // E*8
  unsigned* nmax = (unsigned*)(sbuf + (size_t)E * 8);  // N*8
  float* nsum = (float*)(nmax + (size_t)N * 8);        // N*8
  float* gsum = nsum + (size_t)N * 8;                  // G*288
  float* gvar = gsum + (size_t)G * 288;                // G*288
  float* cnt = gvar + (size_t)G * 288;                 // G
  float* pooled = cnt + G;                             // G*18

  auto zero = [&](float* p, int n) {
    kzero<<<ceilDiv(n, 256), 256, 0, stream>>>(p, n);
  };
  auto gemm = [&](const float* X, const float* W, const float* B, float* Y,
                  int n, int k, int o) {
    dim3 grid(ceilDiv(n, BM), ceilDiv(o, BN));
    gemm_wmma_f32<<<grid, 256, 0, stream>>>(X, W, B, Y, n, k, o);
  };

  // ============ init conv: GATv2(in=9, H=8, C=36, concat) ============
  gemm(x_in, P(4), P(5), S2, N, 9, 288);  // xl
  gemm(x_in, P(6), P(7), S3, N, 9, 288);  // xr
  init_softmax<<<ceilDiv(N * 8, 256), 256, 0, stream>>>(nmax, nsum, N * 8);
  zero(S4, N * 288);
  gat_scores<<<ceilDiv(E * 8, 256), 256, 0, stream>>>(S2, S3, ea, P(8), P(9),
                                                      src, dst, sbuf, nmax, E, 8, 36);
  seg_exp<<<ceilDiv(E * 8, 256), 256, 0, stream>>>(dst, sbuf, nmax, nsum, E, 8);
  gat_msg<<<ceilDiv(E * 288, 256), 256, 0, stream>>>(S2, src, dst, sbuf, nsum,
                                                     S4, E, 8, 36);
  bias_act<<<ceilDiv(N * 288, 256), 256, 0, stream>>>(S4, P(10), N * 288, 288, 0);
  // init graph-norm + elu -> S0
  zero(gsum, G * 288); zero(gvar, G * 288); zero(cnt, G);
  gn_sum<<<ceilDiv(N * 288, 256), 256, 0, stream>>>(S4, batch, gsum, cnt, N, 288);
  gn_center<<<ceilDiv(N * 288, 256), 256, 0, stream>>>(S4, batch, gsum, cnt,
                                                       P(13), S0, gvar, N, 288);
  gn_norm<<<ceilDiv(N * 288, 256), 256, 0, stream>>>(S0, batch, gvar, cnt,
                                                     P(11), P(12), N, 288, 1);

  // ============ 3 layers ============
  for (int i = 0; i < 3; ++i) {
    int base = 14 + 24 * i;
    bool last = (i == 2);
    int H = last ? 1 : 8;
    int C = last ? 18 : 36;
    int HC = H * C;
    float scale = 1.f / sqrtf((float)C);

    // GraphNorm(S0) -> S1  (D = 288 always)
    zero(gsum, G * 288); zero(gvar, G * 288); zero(cnt, G);
    gn_sum<<<ceilDiv(N * 288, 256), 256, 0, stream>>>(S0, batch, gsum, cnt, N, 288);
    gn_center<<<ceilDiv(N * 288, 256), 256, 0, stream>>>(
        S0, batch, gsum, cnt, P(base + 19), S1, gvar, N, 288);
    gn_norm<<<ceilDiv(N * 288, 256), 256, 0, stream>>>(
        S1, batch, gvar, cnt, P(base + 17), P(base + 18), N, 288, 0);

    // ---- GATv2 ----
    gemm(S1, P(base + 0), P(base + 1), S2, N, 288, HC);  // xl
    gemm(S1, P(base + 2), P(base + 3), S3, N, 288, HC);  // xr
    init_softmax<<<ceilDiv(N * H, 256), 256, 0, stream>>>(nmax, nsum, N * H);
    zero(S4, N * HC);
    gat_scores<<<ceilDiv(E * H, 256), 256, 0, stream>>>(
        S2, S3, ea, P(base + 4), P(base + 5), src, dst, sbuf, nmax, E, H, C);
    seg_exp<<<ceilDiv(E * H, 256), 256, 0, stream>>>(dst, sbuf, nmax, nsum, E, H);
    gat_msg<<<ceilDiv(E * HC, 256), 256, 0, stream>>>(S2, src, dst, sbuf, nsum,
                                                      S4, E, H, C);
    bias_act<<<ceilDiv(N * HC, 256), 256, 0, stream>>>(S4, P(base + 6),
                                                       N * HC, HC, 1);  // xg

    // ---- Transformer conv ----
    gemm(S1, P(base + 7), P(base + 8), S2, N, 288, HC);   // q
    gemm(S1, P(base + 9), P(base + 10), S3, N, 288, HC);  // k
    init_softmax<<<ceilDiv(N * H, 256), 256, 0, stream>>>(nmax, nsum, N * H);
    trans_scores<<<ceilDiv(E * H, 256), 256, 0, stream>>>(
        S2, S3, ea, P(base + 13), P(base + 14), src, dst, sbuf, nmax, E, H, C, scale);
    seg_exp<<<ceilDiv(E * H, 256), 256, 0, stream>>>(dst, sbuf, nmax, nsum, E, H);
    gemm(S1, P(base + 11), P(base + 12), S2, N, 288, HC);  // v (overwrites q)
    gemm(S1, P(base + 15), P(base + 16), S5, N, 288, HC);  // skip = xn@Ws^T+bs
    trans_msg<<<ceilDiv(E * HC, 256), 256, 0, stream>>>(
        S2, ea, P(base + 13), P(base + 14), src, dst, sbuf, nsum, S5, E, H, C);

    // x = xg + elu(trans_out) [+ res if same shape]  -> S1
    combine_k<<<ceilDiv(N * HC, 256), 256, 0, stream>>>(
        S4, S5, last ? (const float*)nullptr : S0, S1, N * HC);

    // LN -> S2 ; lin GEMM -> S3 ; x = elu(S3) + S1 -> S0
    ln_k<<<ceilDiv(N, 8), 256, 0, stream>>>(S1, P(base + 20), P(base + 21), S2,
                                            N, HC, 1e-6f);
    gemm(S2, P(base + 22), P(base + 23), S3, N, HC, HC);
    elu_add_k<<<ceilDiv(N * HC, 256), 256, 0, stream>>>(S3, S1, S0, N * HC);
  }

  // ============ global add pool + final head ============
  zero(pooled, G * 18);
  pool_k<<<ceilDiv(N * 18, 256), 256, 0, stream>>>(S0, batch, pooled, N, 18);
  final_k<<<1, 64, 0, stream>>>(pooled, P(86), P(87), P(88), P(89),
                                (float*)d_out);
}